// Ewiser_81037442941605
// MI455X (gfx1250) — compile-verified
//
#include <hip/hip_runtime.h>
#include <hip/hip_bf16.h>

#define Nn   256
#define Dd   1024
#define Cc   50000
#define Gg   4
#define NNZe 300000
#define BK   16
#define BC   80      // C-columns per block: 625 * 80 = 50000 exactly
#define LDA  20      // padded LDS stride: 80B rows (16B aligned), bank-conflict-free
#define SMEM_F ((Nn + BC) * LDA)   // floats per buffer = 6720

typedef __attribute__((ext_vector_type(2))) float v2f;
typedef __attribute__((ext_vector_type(8))) float v8f;

// CDNA5 async global->LDS copy (GVS mode), tracked by ASYNCcnt.
__device__ __forceinline__ void async_b128(unsigned lds, unsigned long long base,
                                           unsigned goff) {
  asm volatile("global_load_async_to_lds_b128 %0, %1, %2"
               :: "v"(lds), "v"(goff), "s"(base) : "memory");
}

// ---------------------------------------------------------------------------
// Kernel 1: BatchNorm (training stats, biased var) -> h0 [N, D]
// ---------------------------------------------------------------------------
__global__ __launch_bounds__(256) void bn_kernel(const float* __restrict__ x,
                                                 float* __restrict__ h0) {
  const int d = blockIdx.x;      // feature column 0..1023
  const int n = threadIdx.x;     // sample row    0..255
  float v = x[(size_t)n * Dd + d];
  __shared__ float s1[256];
  __shared__ float s2[256];
  s1[n] = v;
  s2[n] = v * v;
  __syncthreads();
  for (int s = 128; s > 0; s >>= 1) {
    if (n < s) { s1[n] += s1[n + s]; s2[n] += s2[n + s]; }
    __syncthreads();
  }
  float mean = s1[0] * (1.0f / 256.0f);
  float var  = s2[0] * (1.0f / 256.0f) - mean * mean;
  float rstd = rsqrtf(var + 1e-5f);
  h0[(size_t)n * Dd + d] = (v - mean) * rstd;
}

// ---------------------------------------------------------------------------
// Kernel 2: ZT = swish(W @ h0^T + b) with V_WMMA_F32_16X16X4_F32.
//   A-matrix = W tile (M = c, K = d), B-matrix = h0^T (K = d, N = n).
// Staging: double-buffered GLOBAL_LOAD_ASYNC_TO_LDS_B128 (6 per wave/chunk);
// next chunk's loads are issued before computing the current chunk, then
// s_wait_asynccnt 6 + barrier guarantees residency (async loads complete
// in order).
// ---------------------------------------------------------------------------
__global__ __launch_bounds__(256) void gemm_swish_kernel(
    const float* __restrict__ h0, const float* __restrict__ W,
    const float* __restrict__ bias, float* __restrict__ ZT,
    float* __restrict__ ACC) {
  __shared__ __align__(16) float smem[2 * SMEM_F];  // [h0 256x20 | W 80x20] x2
  __shared__ float Bb[BC];

  const int t    = threadIdx.x;
  const int c0   = blockIdx.x * BC;
  const int wv   = t >> 5;            // wave 0..7
  const int lane = t & 31;
  const int lh   = lane & 15;
  const int ksel = (lane >> 4) << 1;  // 0 (lanes 0-15) or 2 (lanes 16-31)

  if (t < BC) Bb[t] = bias[c0 + t];

  // Runtime scalar LDS base offset (low 32 bits of flat LDS address).
  // NOTE: kept as scalars (no const aggregates) so no addrspacecast constants
  // end up in static initializers.
  const unsigned smemO = (unsigned)(uintptr_t)(void*)smem;
  const unsigned long long h0b = (unsigned long long)(uintptr_t)h0;
  const unsigned long long Wgb = (unsigned long long)(uintptr_t)W;

  // Stage one K-chunk (16 columns) into buffer `buf`: 6 async b128 per thread.
  auto stage = [&](int k0, int buf) {
    const unsigned aO = smemO + (unsigned)buf * (unsigned)(SMEM_F * 4);
    const unsigned wO = aO + (unsigned)(Nn * LDA * 4);
    #pragma unroll
    for (int i = 0; i < 4; ++i) {           // h0: 1024 float4
      int fidx = t + 256 * i;
      int row = fidx >> 2, kq = (fidx & 3) << 2;
      async_b128(aO + (unsigned)(row * LDA + kq) * 4u, h0b,
                 (unsigned)((row * Dd + k0 + kq) * 4));
    }
    #pragma unroll
    for (int i = 0; i < 2; ++i) {           // W: 320 float4 (wrap -> uniform)
      int fidx = t + 256 * i;
      if (fidx >= 320) fidx -= 320;
      int row = fidx >> 2, kq = (fidx & 3) << 2;
      async_b128(wO + (unsigned)(row * LDA + kq) * 4u, Wgb,
                 (unsigned)(((c0 + row) * Dd + k0 + kq) * 4));
    }
  };

  v8f acc[5][2];
  #pragma unroll
  for (int ct = 0; ct < 5; ++ct)
    #pragma unroll
    for (int j = 0; j < 2; ++j) acc[ct][j] = (v8f){};

  stage(0, 0);
  const int NCHUNK = Dd / BK;  // 64
  for (int ki = 0; ki < NCHUNK; ++ki) {
    const int cur = ki & 1;
    if (ki + 1 < NCHUNK) {
      stage((ki + 1) * BK, cur ^ 1);
      asm volatile("s_wait_asynccnt 0x6" ::: "memory");  // chunk ki resident
    } else {
      asm volatile("s_wait_asynccnt 0x0" ::: "memory");
    }
    __syncthreads();  // all waves' loads for chunk ki complete

    const float* Ab = smem + (cur ? SMEM_F : 0);
    const float* Wt = Ab + Nn * LDA;
    #pragma unroll
    for (int kk = 0; kk < BK; kk += 4) {
      const int kb = kk + ksel;  // VGPR0 <- K=kb, VGPR1 <- K=kb+1
      v2f bfrag[2];              // h0^T fragments (wave's 2 n-tiles)
      #pragma unroll
      for (int j = 0; j < 2; ++j)
        bfrag[j] = *(const v2f*)&Ab[(wv * 32 + j * 16 + lh) * LDA + kb];
      v2f afrag[5];              // W fragments (5 c-tiles)
      #pragma unroll
      for (int ct = 0; ct < 5; ++ct)
        afrag[ct] = *(const v2f*)&Wt[(ct * 16 + lh) * LDA + kb];
      #pragma unroll
      for (int ct = 0; ct < 5; ++ct)
        #pragma unroll
        for (int j = 0; j < 2; ++j)
          acc[ct][j] = __builtin_amdgcn_wmma_f32_16x16x4_f32(
              false, afrag[ct], false, bfrag[j], (short)0, acc[ct][j],
              false, false);
    }
    if (ki + 1 < NCHUNK) __syncthreads();  // protect buf before next overwrite
  }

  // Epilogue: bias + swish, store ZT[c][n] and ACC[c][n] (coalesced in n).
  #pragma unroll
  for (int ct = 0; ct < 5; ++ct) {
    #pragma unroll
    for (int j = 0; j < 2; ++j) {
      const int n = wv * 32 + j * 16 + lh;
      #pragma unroll
      for (int v = 0; v < 8; ++v) {
        const int m = v + ((lane >> 4) << 3);  // D layout: lanes16-31 -> M+8
        const int c = c0 + ct * 16 + m;
        float h = acc[ct][j][v] + Bb[ct * 16 + m];
        float z = h / (1.0f + __expf(-h));     // swish
        size_t off = (size_t)c * Nn + n;
        ZT[off]  = z;
        ACC[off] = z;
      }
    }
  }
}

// ---------------------------------------------------------------------------
// Kernel 3: sparse propagation. One wave per edge:
//   ACC[row, :] += (val * vec[g]) * ZT[col, :]   (256 floats, f32 atomics)
// ZT/ACC (102.4 MB) fit in the 192 MB L2 -> atomics stay on-chip.
// ---------------------------------------------------------------------------
__global__ __launch_bounds__(256) void spmm_kernel(
    const float* __restrict__ vals, const float* __restrict__ vec,
    const int* __restrict__ rows, const int* __restrict__ cols,
    const float* __restrict__ ZT, float* __restrict__ ACC) {
  const int wv = threadIdx.x >> 5, lane = threadIdx.x & 31;
  const long long e = (long long)blockIdx.x * 8 + wv;
  if (e >= (long long)Gg * NNZe) return;
  const int g = (int)(e / NNZe);
  const size_t idx = (size_t)e;  // arrays are [G][NNZ] flat
  const float val = vals[idx] * vec[g];
  const int cs = cols[idx];
  const int rd = rows[idx];
  const float4* __restrict__ src = (const float4*)(ZT + (size_t)cs * Nn);
  float* __restrict__ dst = ACC + (size_t)rd * Nn;
  #pragma unroll
  for (int i = 0; i < 2; ++i) {
    float4 m = src[lane + 32 * i];
    int b = (lane + 32 * i) * 4;
    unsafeAtomicAdd(&dst[b + 0], m.x * val);
    unsafeAtomicAdd(&dst[b + 1], m.y * val);
    unsafeAtomicAdd(&dst[b + 2], m.z * val);
    unsafeAtomicAdd(&dst[b + 3], m.w * val);
  }
}

// ---------------------------------------------------------------------------
// Kernel 4: out[n][c] = ACC[c][n]   (LDS-tiled transpose, coalesced both ways)
// ---------------------------------------------------------------------------
__global__ __launch_bounds__(256) void transpose_kernel(
    const float* __restrict__ ACC, float* __restrict__ out) {
  __shared__ float tile[32][33];
  const int cb = blockIdx.x * 32, nb = blockIdx.y * 32;
  const int x = threadIdx.x & 31, y = threadIdx.x >> 5;  // y: 0..7
  #pragma unroll
  for (int i = 0; i < 4; ++i) {
    int c = cb + y + 8 * i;
    if (c < Cc) tile[y + 8 * i][x] = ACC[(size_t)c * Nn + nb + x];
  }
  __syncthreads();
  #pragma unroll
  for (int i = 0; i < 4; ++i) {
    int n = nb + y + 8 * i;
    int c = cb + x;
    if (c < Cc) out[(size_t)n * Cc + c] = tile[x][y + 8 * i];
  }
}

// ---------------------------------------------------------------------------
extern "C" void kernel_launch(void* const* d_in, const int* in_sizes, int n_in,
                              void* d_out, int out_size, void* d_ws,
                              size_t ws_size, hipStream_t stream) {
  (void)in_sizes; (void)n_in; (void)out_size; (void)ws_size;
  const float* output = (const float*)d_in[0];  // [N, D]
  const float* wt2_w  = (const float*)d_in[1];  // [C, D]
  const float* wt2_b  = (const float*)d_in[2];  // [C]
  const float* A_vals = (const float*)d_in[3];  // [G, NNZ]
  const float* vec    = (const float*)d_in[4];  // [G]
  const int*   A_rows = (const int*)d_in[5];    // [G, NNZ]
  const int*   A_cols = (const int*)d_in[6];    // [G, NNZ]
  float* out = (float*)d_out;                   // [N, C]

  // Workspace layout (16B-aligned):
  //   h0  : N*D floats =  1.0 MB
  //   ZT  : C*N floats = 51.2 MB  (Z transposed)
  //   ACC : C*N floats = 51.2 MB  (Z + neighbors, transposed)
  char* ws = (char*)d_ws;
  float* h0  = (float*)ws;
  float* ZT  = (float*)(ws + (size_t)(1 << 20));
  float* ACC = (float*)(ws + (size_t)(1 << 20) + (size_t)Cc * Nn * 4);

  bn_kernel<<<Dd, 256, 0, stream>>>(output, h0);
  gemm_swish_kernel<<<Cc / BC, 256, 0, stream>>>(h0, wt2_w, wt2_b, ZT, ACC);
  spmm_kernel<<<(Gg * NNZe) / 8, 256, 0, stream>>>(A_vals, vec, A_rows, A_cols,
                                                   ZT, ACC);
  transpose_kernel<<<dim3((Cc + 31) / 32, Nn / 32), 256, 0, stream>>>(ACC, out);
}